// S4RKernel_21036749816463
// MI455X (gfx1250) — compile-verified
//
#include <hip/hip_runtime.h>
#include <math.h>

// S4 convolution kernel K[h,l] = 2*Re( sum_n C_eff[h,n] * w[h,n]^l )
// factored as per-h complex 64x64x64 GEMM on the CDNA5 WMMA f32 pipe.

typedef __attribute__((ext_vector_type(2))) float v2f;
typedef __attribute__((ext_vector_type(8))) float v8f;

#define S4_H 256
#define S4_N 64
#define S4_L 4096
#define LDA 65   // padded stride for A-matrix arrays (bank-conflict-free column reads)
#define LDQ 64   // Q reads are row-contiguous across lanes -> no padding needed

__global__ __launch_bounds__(128)
void s4_wmma_kernel(const float* __restrict__ C_re,
                    const float* __restrict__ C_im,
                    const float* __restrict__ log_dt,
                    const float* __restrict__ A_re,
                    const float* __restrict__ A_im,
                    float* __restrict__ Kout) {
    // A-matrix (M=l1, K=n): B[l1][n] = C_eff[n] * w[n]^(64*l1); Ai holds -imag.
    __shared__ float Ar[S4_N * LDA];
    __shared__ float Ai[S4_N * LDA];
    // B-matrix (K=n, N=l0): Q[n][l0] = w[n]^l0
    __shared__ float Qr[S4_N * LDQ];
    __shared__ float Qi[S4_N * LDQ];

    const int h   = blockIdx.x;
    const int tid = threadIdx.x;
    const float dt = __expf(log_dt[h]);

    // ---- Precompute phase: 4096 (n,l) pairs over 128 threads (32 each) ----
    #pragma unroll 4
    for (int i = 0; i < 32; ++i) {
        const int idx = tid * 32 + i;      // 0..4095
        const int n = idx & 63;
        const int l = idx >> 6;
        const float are = A_re[n], aim = A_im[n];
        const float dar = dt * are, dai = dt * aim;   // dtA

        // Q[n][l] = exp(dtA * l)
        {
            const float t = (float)l;
            const float mag = __expf(dar * t);
            float s, c; __sincosf(dai * t, &s, &c);
            Qr[n * LDQ + l] = mag * c;
            Qi[n * LDQ + l] = mag * s;
        }
        // B[l][n] = C[n] * (w - 1)/A * exp(dtA * 64*l),  w = exp(dtA)
        {
            const float t = (float)(l << 6);
            const float mag = __expf(dar * t);
            float s, c; __sincosf(dai * t, &s, &c);
            const float er = mag * c, ei = mag * s;

            const float mw = __expf(dar);
            float sw, cw; __sincosf(dai, &sw, &cw);
            const float wr = mw * cw - 1.0f;           // w - 1 (real)
            const float wi = mw * sw;                  // w - 1 (imag)

            const float inv = 1.0f / (are * are + aim * aim);
            const float dr = (wr * are + wi * aim) * inv;   // (w-1)/A
            const float di = (wi * are - wr * aim) * inv;

            const float cr = C_re[h * S4_N + n], ci = C_im[h * S4_N + n];
            const float br = cr * dr - ci * di;             // C*(w-1)/A
            const float bi = cr * di + ci * dr;

            Ar[l * LDA + n] =   br * er - bi * ei;
            Ai[l * LDA + n] = -(br * ei + bi * er);         // negated: no A/B NEG for f32 WMMA
        }
    }
    __syncthreads();

    // ---- GEMM phase: D[l1][l0] = Ar@Qr + Ai@Qi (Ai pre-negated), K = 2*D ----
    const int wave   = tid >> 5;          // 0..3 -> 16-row strip
    const int lane   = tid & 31;
    const int m_base = wave * 16;
    const int lm     = lane & 15;         // M (A), N (B/D) within tile
    const int kh     = (lane >> 4) << 1;  // K sub-offset: 0 or 2

    float* __restrict__ outp = Kout + (size_t)h * S4_L;

    for (int nb = 0; nb < 64; nb += 16) {
        v8f acc = {};
        #pragma unroll
        for (int k = 0; k < 64; k += 4) {
            v2f a_r, a_i, b_r, b_i;
            const int arow = (m_base + lm) * LDA + k + kh;
            a_r.x = Ar[arow];     a_r.y = Ar[arow + 1];
            a_i.x = Ai[arow];     a_i.y = Ai[arow + 1];
            const int brow = (k + kh) * LDQ + nb + lm;
            b_r.x = Qr[brow];     b_r.y = Qr[brow + LDQ];
            b_i.x = Qi[brow];     b_i.y = Qi[brow + LDQ];
            acc = __builtin_amdgcn_wmma_f32_16x16x4_f32(
                false, a_r, false, b_r, (short)0, acc, false, false);
            acc = __builtin_amdgcn_wmma_f32_16x16x4_f32(
                false, a_i, false, b_i, (short)0, acc, false, false);
        }
        // D layout: VGPR v -> M = v + 8*(lane>=16), N = lane%16
        const int row0 = m_base + ((lane >> 4) << 3);
        const int col  = nb + lm;
        #pragma unroll
        for (int v = 0; v < 8; ++v) {
            outp[(row0 + v) * 64 + col] = 2.0f * acc[v];
        }
    }
}

extern "C" void kernel_launch(void* const* d_in, const int* in_sizes, int n_in,
                              void* d_out, int out_size, void* d_ws, size_t ws_size,
                              hipStream_t stream) {
    const float* C_re   = (const float*)d_in[0];   // (H, N)
    const float* C_im   = (const float*)d_in[1];   // (H, N)
    const float* log_dt = (const float*)d_in[2];   // (H,)
    const float* A_re   = (const float*)d_in[3];   // (N,)
    const float* A_im   = (const float*)d_in[4];   // (N,)
    // d_in[5] = input_length (fixed 4096)
    float* K = (float*)d_out;                      // (H, L)

    s4_wmma_kernel<<<S4_H, 128, 0, stream>>>(C_re, C_im, log_dt, A_re, A_im, K);
}